// LengthRegulator_55825984914003
// MI455X (gfx1250) — compile-verified
//
#include <hip/hip_runtime.h>
#include <stdint.h>

// Problem shape (static in the reference)
#define B_SZ      16
#define T_IN      512
#define D_SZ      384              // floats per row = 1536 bytes
#define MAX_LEN_C 2048
#define WAVE      32
#define FRAMES_PER_WAVE 16         // TDM gather handles up to 16 row indices
#define WAVES_PB  2                // 2 waves/block -> 48 KB LDS/block

typedef __attribute__((ext_vector_type(4))) unsigned uint4v;
typedef __attribute__((ext_vector_type(8))) unsigned uint8v;

// ---------------------------------------------------------------------------
// Tensor DMA descriptor builders (CDNA5 ISA)
// ---------------------------------------------------------------------------
__device__ __forceinline__ uint4v tdm_group0(unsigned flags, unsigned lds_addr,
                                             uint64_t gaddr) {
  uint4v g;
  g[0] = flags;                                   // count=1 [, gather bits]
  g[1] = lds_addr;                                // lds_addr[31:0]
  g[2] = (unsigned)gaddr;                         // global_addr[31:0]
  g[3] = ((unsigned)(gaddr >> 32) & 0x01FFFFFFu)  // global_addr[56:32]
         | 0x80000000u;                           // type = 2 ("image")
  return g;
}

__device__ __forceinline__ uint8v tdm_group1(unsigned dim0, unsigned dim1,
                                             unsigned tile0, unsigned tile1,
                                             unsigned stride0) {
  uint8v g;
  g[0] = 0x00020000u;                             // wg_mask=0, data_size=4B
  g[1] = (dim0 & 0xFFFFu) << 16;                  // tensor_dim0[15:0] @63:48
  g[2] = (dim0 >> 16) | ((dim1 & 0xFFFFu) << 16); // dim0[31:16], dim1[15:0]
  g[3] = (dim1 >> 16) | (tile0 << 16);            // dim1[31:16], tile_dim0
  g[4] = tile1 & 0xFFFFu;                         // tile_dim1 (tile_dim2=0)
  g[5] = stride0;                                 // tensor_dim0_stride[31:0]
  g[6] = 0;                                       // stride0[47:32], stride1 lo
  g[7] = 0;                                       // stride1 hi
  return g;
}

// ---------------------------------------------------------------------------
// Kernel 1: per-batch inclusive scan of reps -> cum[b][0..T_IN], dec_lens[b]
// ---------------------------------------------------------------------------
__global__ __launch_bounds__(T_IN)
void lr_scan_kernel(const int* __restrict__ dur,
                    const int* __restrict__ max_len_p,
                    int* __restrict__ cum,
                    int* __restrict__ dec_lens_out) {
  __shared__ int s[T_IN + 1];
  const int b   = blockIdx.x;
  const int tid = threadIdx.x;

  int d   = dur[b * T_IN + tid];
  int rep = (int)floorf((float)d + 0.5f);         // reps = floor(dur + 0.5)
  s[tid + 1] = rep;
  if (tid == 0) s[0] = 0;
  __syncthreads();

  for (int off = 1; off < T_IN; off <<= 1) {      // Hillis-Steele scan
    int t = (tid >= off) ? s[tid + 1 - off] : 0;
    __syncthreads();
    if (tid >= off) s[tid + 1] += t;
    __syncthreads();
  }

  cum[b * (T_IN + 1) + tid] = s[tid];
  if (tid == 0) {
    int total = s[T_IN];
    cum[b * (T_IN + 1) + T_IN] = total;
    int ml = max_len_p[0];
    dec_lens_out[b] = total < ml ? total : ml;
  }
}

// ---------------------------------------------------------------------------
// Kernel 2: one wave32 per 16 consecutive output frames.
//   lanes 0..15 binary-search cum[b] -> monotone row indices (cumsum => the
//   gather-mode "non-decreasing index" requirement holds by construction)
//   TDM gather load: 16 rows of x[b] -> LDS          (1 instruction, 24 KB)
//   TDM store:       contiguous 6144 floats -> out   (1 instruction, 24 KB)
//   padded frames are a suffix: tile_dim1=nvalid + zero-filled LDS suffix
// ---------------------------------------------------------------------------
__global__ __launch_bounds__(WAVE * WAVES_PB)
void lr_tdm_gather_kernel(const float* __restrict__ x,
                          const int* __restrict__ cum,
                          float* __restrict__ out) {
  __shared__ float stage[WAVES_PB * FRAMES_PER_WAVE * D_SZ];   // 48 KB

  const int lane = threadIdx.x & (WAVE - 1);
  const int wave = threadIdx.x >> 5;
  const int wg   = blockIdx.x * WAVES_PB + wave;       // global wave id
  const int f0   = wg * FRAMES_PER_WAVE;               // first frame (global)
  const int b    = f0 >> 11;                           // / MAX_LEN_C
  const int l0   = f0 & (MAX_LEN_C - 1);

  const int* crow  = cum + b * (T_IN + 1);
  const int  total = crow[T_IN];

  int nvalid = total - l0;
  nvalid = nvalid < 0 ? 0 : (nvalid > FRAMES_PER_WAVE ? FRAMES_PER_WAVE : nvalid);

  // per-lane binary search: largest i with crow[i] <= l  (lanes 0..15)
  const int l = l0 + (lane < FRAMES_PER_WAVE ? lane : FRAMES_PER_WAVE - 1);
  int lo = 0, hi = T_IN;
  while (hi - lo > 1) {
    int mid = (lo + hi) >> 1;
    if (crow[mid] <= l) lo = mid; else hi = mid;
  }

  // pack 16 x 16-bit row indices into D# groups 2/3 (wave-uniform via readlane)
  unsigned p[8];
#pragma unroll
  for (int k = 0; k < 8; ++k) {
    unsigned e = (unsigned)__builtin_amdgcn_readlane(lo, 2 * k)     & 0xFFFFu;
    unsigned o = (unsigned)__builtin_amdgcn_readlane(lo, 2 * k + 1) & 0xFFFFu;
    p[k] = e | (o << 16);
  }
  uint4v gi2 = { p[0], p[1], p[2], p[3] };
  uint4v gi3 = { p[4], p[5], p[6], p[7] };

  float*   slab     = &stage[wave * FRAMES_PER_WAVE * D_SZ];
  unsigned lds_base = (unsigned)(uintptr_t)slab;

  // zero the LDS suffix for padded frames (l >= total); DScnt tracked
  float4* s4 = (float4*)slab;
#pragma unroll 1
  for (int u = nvalid * (D_SZ / 4) + lane; u < FRAMES_PER_WAVE * (D_SZ / 4); u += WAVE)
    s4[u] = make_float4(0.f, 0.f, 0.f, 0.f);

  // ---- TDM gather load: x[b, idx[k], :] -> LDS slot k (TENSORcnt) ----
  uint4v g0 = tdm_group0(0x80000001u,              // count=1, gather, 16b idx
                         lds_base,
                         (uint64_t)(uintptr_t)(x + (size_t)b * T_IN * D_SZ));
  uint8v g1 = tdm_group1(/*dim0*/ D_SZ, /*dim1*/ T_IN,
                         /*tile0*/ D_SZ, /*tile1=#indices*/ (unsigned)nvalid,
                         /*stride0*/ D_SZ);
  asm volatile("tensor_load_to_lds %0, %1, %2, %3"
               :: "s"(g0), "s"(g1), "s"(gi2), "s"(gi3) : "memory");

  // LDS must be fully written (zero suffix + gathered rows) before TDM store
  asm volatile("s_wait_dscnt 0" ::: "memory");
  __builtin_amdgcn_s_wait_tensorcnt(0);

  // ---- TDM store: contiguous 16*384 floats LDS -> out[f0*D_SZ] ----
  uint4v z4 = { 0u, 0u, 0u, 0u };
  uint4v h0 = tdm_group0(0x00000001u, lds_base,
                         (uint64_t)(uintptr_t)(out + (size_t)f0 * D_SZ));
  uint8v h1 = tdm_group1(/*dim0*/ FRAMES_PER_WAVE * D_SZ, /*dim1*/ 1,
                         /*tile0*/ FRAMES_PER_WAVE * D_SZ, /*tile1*/ 1,
                         /*stride0*/ FRAMES_PER_WAVE * D_SZ);
  asm volatile("tensor_store_from_lds %0, %1, %2, %3"
               :: "s"(h0), "s"(h1), "s"(z4), "s"(z4) : "memory");

  __builtin_amdgcn_s_wait_tensorcnt(0);   // s_endpgm also waits idle
}

// ---------------------------------------------------------------------------
extern "C" void kernel_launch(void* const* d_in, const int* in_sizes, int n_in,
                              void* d_out, int out_size, void* d_ws, size_t ws_size,
                              hipStream_t stream) {
  const float* x         = (const float*)d_in[0];
  const int*   durations = (const int*)d_in[1];
  const int*   max_len_p = (const int*)d_in[2];

  float* out      = (float*)d_out;
  int*   dec_lens = (int*)((float*)d_out + (size_t)B_SZ * MAX_LEN_C * D_SZ);
  int*   cum      = (int*)d_ws;   // B * (T_IN+1) ints = 32.8 KB

  lr_scan_kernel<<<B_SZ, T_IN, 0, stream>>>(durations, max_len_p, cum, dec_lens);

  const int nframes = B_SZ * MAX_LEN_C;                       // 32768
  const int nblocks = nframes / (FRAMES_PER_WAVE * WAVES_PB); // 1024
  lr_tdm_gather_kernel<<<nblocks, WAVE * WAVES_PB, 0, stream>>>(x, cum, out);
}